// KnowledgeGuidedTransform_75213467287748
// MI455X (gfx1250) — compile-verified
//
#include <hip/hip_runtime.h>
#include <hip/hip_bf16.h>

typedef float v2f __attribute__((ext_vector_type(2)));
typedef float v8f __attribute__((ext_vector_type(8)));

#define NF   256      // FL == FA == FO
#define CC   768      // concept dim
#define NE   128      // EL == EA == EO
#define NORG 16
#define NABN 64
#define LPAD 260      // padded stride for organ/comb buffers (floats)
#define XST  36       // padded stride for staged x chunks (floats), 64 rows x 32 k
#define XBUF 2304     // 64 * XST floats per staging buffer

// ---------------------------------------------------------------------------
// Kernel 1a: degree counts (tiny)
// ---------------------------------------------------------------------------
__global__ __launch_bounds__(128)
void kg_deg(const int* __restrict__ lab_org, const int* __restrict__ abn_org,
            const int* __restrict__ o2a_a,
            float* __restrict__ degL, float* __restrict__ degA, float* __restrict__ degO)
{
    int t = threadIdx.x;
    if (t < 16) {
        int c = 0;
        for (int e = 0; e < NE; ++e) c += (lab_org[e] == t);
        degL[t] = (float)c;
    } else if (t < 32) {
        int o = t - 16, c = 0;
        for (int e = 0; e < NE; ++e) c += (abn_org[e] == o);
        degA[o] = (float)c;
    } else if (t < 96) {
        int a = t - 32, c = 0;
        for (int e = 0; e < NE; ++e) c += (o2a_a[e] == a);
        degO[a] = fmaxf((float)c, 1.0f);   // reference: deg = max(count, 1)
    }
}

// ---------------------------------------------------------------------------
// Kernel 1b: batch-invariant precompute
//   which==0: cb_lab[e,g] = (lab_ce[lab_idx[e]] + lab_re[e]) @ W_lab[NF:] + b_lab
//   which==1: cb_abn[e,g] = (abn_ce[abn_idx[e]] + abn_re[e]) @ W_abn[NF:] + b_abn
//   which==2: dd[e,g]     =  lab_re[e] @ D
// ---------------------------------------------------------------------------
__global__ __launch_bounds__(256)
void kg_pre(const float* __restrict__ lab_ce, const float* __restrict__ abn_ce,
            const float* __restrict__ lab_re, const float* __restrict__ abn_re,
            const float* __restrict__ W_lab,  const float* __restrict__ b_lab,
            const float* __restrict__ W_abn,  const float* __restrict__ b_abn,
            const float* __restrict__ Dm,
            const int* __restrict__ lab_idx,  const int* __restrict__ abn_idx,
            float* __restrict__ cb_lab, float* __restrict__ cb_abn, float* __restrict__ dd)
{
    __shared__ float crow[CC];
    const int e = blockIdx.x;
    const int which = blockIdx.y;
    const int g = threadIdx.x;

    for (int k = g; k < CC; k += 256) {
        float v;
        if (which == 0)      v = lab_re[e*CC + k] + lab_ce[lab_idx[e]*CC + k];
        else if (which == 1) v = abn_re[e*CC + k] + abn_ce[abn_idx[e]*CC + k];
        else                 v = lab_re[e*CC + k];
        crow[k] = v;
    }
    __syncthreads();

    const float* W2 = (which == 0) ? (W_lab + (size_t)NF*NF)
                    : (which == 1) ? (W_abn + (size_t)NF*NF) : Dm;
    float acc = (which == 0) ? b_lab[g] : (which == 1) ? b_abn[g] : 0.0f;
    #pragma unroll 8
    for (int k = 0; k < CC; ++k) acc += crow[k] * W2[(size_t)k*NF + g];

    float* dst = (which == 0) ? cb_lab : (which == 1) ? cb_abn : dd;
    dst[e*NF + g] = acc;
}

// ---------------------------------------------------------------------------
// Async gather-stage: 64 edge rows x 32 floats -> LDS (CDNA5 async-to-LDS DMA)
// 256 threads: 4 threads/row, 2x b128 each. ASYNCcnt tracked.
// ---------------------------------------------------------------------------
__device__ __forceinline__
void stage_async(const float* __restrict__ feats, const int* __restrict__ eidx,
                 int bt, int mh, int kc, uint32_t xs_byte, int tid)
{
    const int r = tid >> 2;                       // 0..63 local row
    const int q = tid & 3;                        // 8-float sub-chunk
    const int edge = mh * 64 + r;
    const float* gsrc = feats + ((size_t)bt * 64 + eidx[edge]) * NF + kc * 32 + q * 8;
    uint32_t l = xs_byte + (uint32_t)(r * XST + q * 8) * 4u;
    uint64_t g = (uint64_t)(uintptr_t)gsrc;
    asm volatile("global_load_async_to_lds_b128 %0, %1, off" :: "v"(l), "v"(g) : "memory");
    l += 16u; g += 16u;
    asm volatile("global_load_async_to_lds_b128 %0, %1, off" :: "v"(l), "v"(g) : "memory");
}

// ---------------------------------------------------------------------------
// Kernel 2: fused per-(b,t) pipeline. One 256-thread (8-wave) WG per (b,t).
// ---------------------------------------------------------------------------
__global__ __launch_bounds__(256)
void kg_main(const float* __restrict__ lab_feats, const float* __restrict__ abn_feats,
             const float* __restrict__ W_lab, const float* __restrict__ W_abn,
             const float* __restrict__ W_org, const float* __restrict__ b_org,
             const float* __restrict__ W_o2a, const float* __restrict__ b_o2a,
             const unsigned char* __restrict__ time_mask,
             const int* __restrict__ lab_idx, const int* __restrict__ lab_org_idx,
             const int* __restrict__ abn_idx, const int* __restrict__ abn_org_idx,
             const int* __restrict__ o2a_abn_idx, const int* __restrict__ o2a_org_idx,
             const float* __restrict__ cb_lab, const float* __restrict__ cb_abn,
             const float* __restrict__ dd,
             const float* __restrict__ degL, const float* __restrict__ degA,
             const float* __restrict__ degO,
             float* __restrict__ out)
{
    __shared__ float xs [2 * XBUF];    // double-buffered staged x chunks; aliased as comb (16xLPAD)
    __shared__ float os1[16 * LPAD];   // lab organ sums -> organ_states
    __shared__ float os2[16 * LPAD];   // abn organ sums -> msg
    __shared__ int   e_lidx[NE], e_lorg[NE], e_aidx[NE], e_aorg[NE], e_oa[NE], e_oo[NE];
    __shared__ float orgmask[NORG];

    const int bt   = blockIdx.x;           // 0..511
    const int tid  = threadIdx.x;
    const int lane = tid & 31;
    const int wave = tid >> 5;             // 0..7
    const int col  = lane & 15;            // N within tile / M row for A
    const int hi   = lane >> 4;            // lane half
    const int koff = hi * 2;               // K sub-offset for A/B fragments
    const uint32_t xs_byte = (uint32_t)(uintptr_t)xs;   // wave-relative LDS byte offset

    for (int i = tid; i < NE; i += 256) {
        e_lidx[i] = lab_idx[i];     e_lorg[i] = lab_org_idx[i];
        e_aidx[i] = abn_idx[i];     e_aorg[i] = abn_org_idx[i];
        e_oa[i]   = o2a_abn_idx[i]; e_oo[i]   = o2a_org_idx[i];
    }
    for (int i = tid; i < 16 * LPAD; i += 256) { os1[i] = 0.0f; os2[i] = 0.0f; }
    __syncthreads();

    // ======= Phase A/B: edge GEMMs (lab then abn), 2 M-halves each.
    // Per (path, half): K swept in 8 chunks of 32; x chunk async-staged to LDS
    // (double buffered), B fragments held in registers across 4 M-tiles.
    for (int path = 0; path < 2; ++path) {
        const float* feats = (path == 0) ? lab_feats : abn_feats;
        const float* Wm    = (path == 0) ? W_lab     : W_abn;     // rows [0, NF)
        const float* cb    = (path == 0) ? cb_lab    : cb_abn;
        const int*   eidx  = (path == 0) ? e_lidx    : e_aidx;
        const int*   eorg  = (path == 0) ? e_lorg    : e_aorg;
        float*       osum  = (path == 0) ? os1       : os2;

        for (int mh = 0; mh < 2; ++mh) {
            v8f acc[4][2] = {};                       // 4 M-tiles x 2 N-tiles
            const int n0 = wave * 32;

            stage_async(feats, eidx, bt, mh, 0, xs_byte, tid);   // prologue chunk 0

            for (int kc = 0; kc < 8; ++kc) {
                asm volatile("s_wait_asynccnt 0x0" ::: "memory");
                __syncthreads();                      // chunk kc visible; prev buffer free
                if (kc + 1 < 8)
                    stage_async(feats, eidx, bt, mh, kc + 1,
                                xs_byte + (uint32_t)(((kc + 1) & 1) * XBUF * 4), tid);

                // ---- B fragments for this K-chunk (held in VGPRs, reused by 4 M-tiles)
                v2f bf0[8], bf1[8];
                const float* bp0 = Wm + (size_t)(kc * 32 + koff) * NF + n0 + col;
                const float* bp1 = bp0 + 16;
                #pragma unroll
                for (int j = 0; j < 8; ++j) {
                    bf0[j].x = bp0[(size_t)(j * 4) * NF];
                    bf0[j].y = bp0[(size_t)(j * 4 + 1) * NF];
                    bf1[j].x = bp1[(size_t)(j * 4) * NF];
                    bf1[j].y = bp1[(size_t)(j * 4 + 1) * NF];
                }

                // ---- 64 WMMAs against the staged chunk
                const float* ap = xs + (kc & 1) * XBUF + col * XST + koff;
                #pragma unroll
                for (int mt = 0; mt < 4; ++mt) {
                    #pragma unroll
                    for (int j = 0; j < 8; ++j) {
                        v2f a;
                        a.x = ap[mt * 16 * XST + j * 4];
                        a.y = ap[mt * 16 * XST + j * 4 + 1];
                        acc[mt][0] = __builtin_amdgcn_wmma_f32_16x16x4_f32(false, a, false, bf0[j], (short)0, acc[mt][0], false, false);
                        acc[mt][1] = __builtin_amdgcn_wmma_f32_16x16x4_f32(false, a, false, bf1[j], (short)0, acc[mt][1], false, false);
                    }
                }
            }

            // ---- epilogue: h = tanh(acc + cbias) (+ rel@D for lab); organ scatter
            #pragma unroll
            for (int mt = 0; mt < 4; ++mt) {
                #pragma unroll
                for (int r = 0; r < 8; ++r) {
                    const int m    = r + 8 * hi;          // C/D row mapping
                    const int edge = mh * 64 + mt * 16 + m;
                    const int o    = eorg[edge];
                    const int n    = n0 + col;
                    float v0 = tanhf(acc[mt][0][r] + cb[edge * NF + n]);
                    float v1 = tanhf(acc[mt][1][r] + cb[edge * NF + n + 16]);
                    if (path == 0) {
                        v0 += dd[edge * NF + n];
                        v1 += dd[edge * NF + n + 16];
                    }
                    atomicAdd(&osum[o * LPAD + n],      v0);   // ds_add_f32
                    atomicAdd(&osum[o * LPAD + n + 16], v1);
                }
            }
        }
    }
    __syncthreads();   // all organ scatters complete

    // ======= Phase C1: comb = mean_lab + mean_abn (into xs alias), organ mask
    const float wmask = time_mask[bt] ? 1.0f : 0.0f;
    for (int i = tid; i < NORG * NF; i += 256) {
        const int o = i >> 8, g = i & (NF - 1);
        const float cl = wmask * degL[o];
        const float ca = wmask * degA[o];
        const float ml = os1[o * LPAD + g] / fmaxf(cl, 1e-9f);
        const float ma = os2[o * LPAD + g] / fmaxf(ca, 1e-9f);
        xs[o * LPAD + g] = ml + ma;
        if (g == 0) orgmask[o] = (cl > 0.0f || ca > 0.0f) ? 1.0f : 0.0f;
    }
    __syncthreads();

    // ======= Phase C2: organ_states = tanh(comb @ W_org + b_org) * mask  -> os1
    {
        v8f acc0 = {}; v8f acc1 = {};
        const int n0 = wave * 32;
        const float* aptr = xs + col * LPAD + koff;
        const float* b0p  = W_org + (size_t)koff * NF + n0 + col;
        const float* b1p  = b0p + 16;
        #pragma unroll 8
        for (int k0 = 0; k0 < NF; k0 += 4) {
            v2f a;  a.x  = aptr[k0];             a.y  = aptr[k0 + 1];
            v2f b0; b0.x = b0p[(size_t)k0 * NF]; b0.y = b0p[(size_t)(k0 + 1) * NF];
            v2f b1; b1.x = b1p[(size_t)k0 * NF]; b1.y = b1p[(size_t)(k0 + 1) * NF];
            acc0 = __builtin_amdgcn_wmma_f32_16x16x4_f32(false, a, false, b0, (short)0, acc0, false, false);
            acc1 = __builtin_amdgcn_wmma_f32_16x16x4_f32(false, a, false, b1, (short)0, acc1, false, false);
        }
        __syncthreads();   // done reading comb / old os1
        #pragma unroll
        for (int r = 0; r < 8; ++r) {
            const int m = r + 8 * hi;
            const int n = n0 + col;
            os1[m * LPAD + n]      = tanhf(acc0[r] + b_org[n])      * orgmask[m];
            os1[m * LPAD + n + 16] = tanhf(acc1[r] + b_org[n + 16]) * orgmask[m];
        }
    }
    __syncthreads();

    // ======= Phase C3: msg = organ_states @ W_o2a + b_o2a  -> os2
    {
        v8f acc0 = {}; v8f acc1 = {};
        const int n0 = wave * 32;
        const float* aptr = os1 + col * LPAD + koff;
        const float* b0p  = W_o2a + (size_t)koff * NF + n0 + col;
        const float* b1p  = b0p + 16;
        #pragma unroll 8
        for (int k0 = 0; k0 < NF; k0 += 4) {
            v2f a;  a.x  = aptr[k0];             a.y  = aptr[k0 + 1];
            v2f b0; b0.x = b0p[(size_t)k0 * NF]; b0.y = b0p[(size_t)(k0 + 1) * NF];
            v2f b1; b1.x = b1p[(size_t)k0 * NF]; b1.y = b1p[(size_t)(k0 + 1) * NF];
            acc0 = __builtin_amdgcn_wmma_f32_16x16x4_f32(false, a, false, b0, (short)0, acc0, false, false);
            acc1 = __builtin_amdgcn_wmma_f32_16x16x4_f32(false, a, false, b1, (short)0, acc1, false, false);
        }
        __syncthreads();   // done reading old os2
        #pragma unroll
        for (int r = 0; r < 8; ++r) {
            const int m = r + 8 * hi;
            const int n = n0 + col;
            os2[m * LPAD + n]      = acc0[r] + b_o2a[n];
            os2[m * LPAD + n + 16] = acc1[r] + b_o2a[n + 16];
        }
    }
    __syncthreads();

    // ======= Phase D: abn_out = abn_feats + (scatter-mean of gathered msg)
    {
        const int a  = tid >> 2;              // 0..63 abnormal slot
        const int g0 = (tid & 3) * 64;        // 64-float column chunk
        const float invdeg = 1.0f / degO[a];
        const float4* fin  = (const float4*)(abn_feats + ((size_t)bt * NABN + a) * NF + g0);
        float4*       fout = (float4*)(out       + ((size_t)bt * NABN + a) * NF + g0);

        float4 accv[16];
        #pragma unroll
        for (int j = 0; j < 16; ++j) accv[j] = make_float4(0.f, 0.f, 0.f, 0.f);

        for (int e = 0; e < NE; ++e) {
            if (e_oa[e] == a) {
                const float* mrow = os2 + e_oo[e] * LPAD + g0;
                #pragma unroll
                for (int j = 0; j < 16; ++j) {
                    accv[j].x += mrow[4 * j + 0];
                    accv[j].y += mrow[4 * j + 1];
                    accv[j].z += mrow[4 * j + 2];
                    accv[j].w += mrow[4 * j + 3];
                }
            }
        }
        #pragma unroll
        for (int j = 0; j < 16; ++j) {
            float4 f = fin[j];
            float4 o4;
            o4.x = f.x + accv[j].x * invdeg;
            o4.y = f.y + accv[j].y * invdeg;
            o4.z = f.z + accv[j].z * invdeg;
            o4.w = f.w + accv[j].w * invdeg;
            fout[j] = o4;
        }
    }
}

// ---------------------------------------------------------------------------
extern "C" void kernel_launch(void* const* d_in, const int* in_sizes, int n_in,
                              void* d_out, int out_size, void* d_ws, size_t ws_size,
                              hipStream_t stream)
{
    const float* lab_feats = (const float*)d_in[0];
    const float* abn_feats = (const float*)d_in[1];
    const float* lab_ce    = (const float*)d_in[2];
    const float* abn_ce    = (const float*)d_in[3];
    const float* lab_re    = (const float*)d_in[4];
    const float* abn_re    = (const float*)d_in[5];
    const float* W_lab     = (const float*)d_in[6];
    const float* b_lab     = (const float*)d_in[7];
    const float* W_abn     = (const float*)d_in[8];
    const float* b_abn     = (const float*)d_in[9];
    const float* W_org     = (const float*)d_in[10];
    const float* b_org     = (const float*)d_in[11];
    const float* Dm        = (const float*)d_in[12];
    const float* W_o2a     = (const float*)d_in[13];
    const float* b_o2a     = (const float*)d_in[14];
    const unsigned char* tmask = (const unsigned char*)d_in[15];
    const int* lab_idx     = (const int*)d_in[16];
    const int* lab_org     = (const int*)d_in[17];
    const int* abn_idx     = (const int*)d_in[18];
    const int* abn_org     = (const int*)d_in[19];
    const int* o2a_a       = (const int*)d_in[20];
    const int* o2a_o       = (const int*)d_in[21];

    float* ws     = (float*)d_ws;
    float* cb_lab = ws;                 // 128*256
    float* cb_abn = ws + 32768;         // 128*256
    float* dd     = ws + 65536;         // 128*256
    float* degL   = ws + 98304;         // 16
    float* degA   = ws + 98320;         // 16
    float* degO   = ws + 98336;         // 64

    kg_deg<<<1, 128, 0, stream>>>(lab_org, abn_org, o2a_a, degL, degA, degO);
    kg_pre<<<dim3(128, 3), 256, 0, stream>>>(lab_ce, abn_ce, lab_re, abn_re,
                                             W_lab, b_lab, W_abn, b_abn, Dm,
                                             lab_idx, abn_idx, cb_lab, cb_abn, dd);
    kg_main<<<512, 256, 0, stream>>>(lab_feats, abn_feats, W_lab, W_abn,
                                     W_org, b_org, W_o2a, b_o2a, tmask,
                                     lab_idx, lab_org, abn_idx, abn_org, o2a_a, o2a_o,
                                     cb_lab, cb_abn, dd, degL, degA, degO,
                                     (float*)d_out);
}